// HAN_67276367724658
// MI455X (gfx1250) — compile-verified
//
#include <hip/hip_runtime.h>

// ---------------------------------------------------------------------------
// HAN (3-layer heterogeneous graph attention) + set-transformer readout for
// MI455X / gfx1250.  GEMMs use V_WMMA_F32_16X16X4_F32 (fp32-exact matrix
// path) with 16x64 tiles per wave; edge softmax/aggregation uses L2 atomics
// (feature tables fit in the 192MB L2).
//
// Input flattening assumption: setup_inputs() dict insertion order for the 8
// tensor/edge entries, then `params` flattened as a JAX pytree (dict keys
// sorted alphabetically at every nesting level) -> 81 flat device inputs.
// ---------------------------------------------------------------------------

typedef __attribute__((ext_vector_type(2))) float v2f;
typedef __attribute__((ext_vector_type(8))) float v8f;

#define SEG_EPS 1e-16f

static __device__ __forceinline__ unsigned fenc(float x) {
  unsigned b = __float_as_uint(x);
  return (b & 0x80000000u) ? ~b : (b | 0x80000000u);
}
static __device__ __forceinline__ float fdec(unsigned u) {
  unsigned b = (u & 0x80000000u) ? (u & 0x7FFFFFFFu) : ~u;
  return __uint_as_float(b);
}

// ---------------------------------------------------------------------------
// WMMA fp32 GEMM:  C[M,N] = act(A[M,K] @ B[K,N] + bias[N])
// One wave32 per 16x64 strip (4 x 16x16 tiles share the A fragment).
// A 16x4 layout: lane l -> m=l&15, k pair (l>>4)*2.  B 4x16: lane l ->
// n=l&15, same k pair.  D: VGPR j -> row j + 8*(l>>4), col l&15.
// Requires: K % 16 == 0, N % 64 == 0 (true for every call here).
// ---------------------------------------------------------------------------
__global__ __launch_bounds__(128) void gemm_wmma(
    const float* __restrict__ A, const float* __restrict__ B,
    const float* __restrict__ bias, float* __restrict__ C,
    int M, int N, int K, int act) {
  const int lane = threadIdx.x;
  const int m0 = (blockIdx.y * 4 + threadIdx.y) * 16;
  if (m0 >= M) return;  // uniform per wave
  const int n0 = blockIdx.x * 64;
  const int mh = lane & 15;
  const int kk = (lane >> 4) << 1;
  const int rowA = m0 + mh;
  const float mask = (rowA < M) ? 1.f : 0.f;  // branchless row guard
  const float* Ap = A + (size_t)(rowA < M ? rowA : 0) * K + kk;
  const float* Bp = B + (size_t)kk * N + (n0 + mh);
  const size_t bstep = (size_t)4 * N;
  v8f acc0 = {0.f, 0.f, 0.f, 0.f, 0.f, 0.f, 0.f, 0.f};
  v8f acc1 = acc0, acc2 = acc0, acc3 = acc0;
#pragma unroll 4
  for (int k0 = 0; k0 < K; k0 += 4) {
    v2f a = *(const v2f*)Ap;          // global_load_b64, unconditional
    a.x *= mask;
    a.y *= mask;
    v2f b0, b1, b2, b3;
    b0.x = Bp[0];
    b0.y = Bp[(size_t)N];
    b1.x = Bp[16];
    b1.y = Bp[(size_t)N + 16];
    b2.x = Bp[32];
    b2.y = Bp[(size_t)N + 32];
    b3.x = Bp[48];
    b3.y = Bp[(size_t)N + 48];
    acc0 = __builtin_amdgcn_wmma_f32_16x16x4_f32(false, a, false, b0, (short)0,
                                                 acc0, false, false);
    acc1 = __builtin_amdgcn_wmma_f32_16x16x4_f32(false, a, false, b1, (short)0,
                                                 acc1, false, false);
    acc2 = __builtin_amdgcn_wmma_f32_16x16x4_f32(false, a, false, b2, (short)0,
                                                 acc2, false, false);
    acc3 = __builtin_amdgcn_wmma_f32_16x16x4_f32(false, a, false, b3, (short)0,
                                                 acc3, false, false);
    Ap += 4;
    Bp += bstep;
  }
  const int mb = m0 + ((lane >> 4) << 3);
  v8f accs[4] = {acc0, acc1, acc2, acc3};
#pragma unroll
  for (int t = 0; t < 4; ++t) {
    const int n = n0 + 16 * t + mh;
    const float bv = bias ? bias[n] : 0.f;
#pragma unroll
    for (int j = 0; j < 8; ++j) {
      int m = mb + j;
      if (m < M) {
        float v = accs[t][j] + bv;
        if (act == 1) v = fmaxf(v, 0.f);
        C[(size_t)m * N + n] = v;
      }
    }
  }
}

// Same GEMM, but fused semantic-attention epilogue:
//   score += sum_{m,n in strip} q[n] * tanh(acc + kb[n])
// Wave-level shuffle reduction, one atomicAdd per wave.
__global__ __launch_bounds__(128) void gemm_wmma_score(
    const float* __restrict__ A, const float* __restrict__ B,
    const float* __restrict__ kb, const float* __restrict__ q,
    float* __restrict__ score, int M, int N, int K) {
  const int lane = threadIdx.x;
  const int m0 = (blockIdx.y * 4 + threadIdx.y) * 16;
  if (m0 >= M) return;
  const int n0 = blockIdx.x * 64;
  const int mh = lane & 15;
  const int kk = (lane >> 4) << 1;
  const int rowA = m0 + mh;
  const float mask = (rowA < M) ? 1.f : 0.f;
  const float* Ap = A + (size_t)(rowA < M ? rowA : 0) * K + kk;
  const float* Bp = B + (size_t)kk * N + (n0 + mh);
  const size_t bstep = (size_t)4 * N;
  v8f acc0 = {0.f, 0.f, 0.f, 0.f, 0.f, 0.f, 0.f, 0.f};
  v8f acc1 = acc0, acc2 = acc0, acc3 = acc0;
#pragma unroll 4
  for (int k0 = 0; k0 < K; k0 += 4) {
    v2f a = *(const v2f*)Ap;
    a.x *= mask;
    a.y *= mask;
    v2f b0, b1, b2, b3;
    b0.x = Bp[0];
    b0.y = Bp[(size_t)N];
    b1.x = Bp[16];
    b1.y = Bp[(size_t)N + 16];
    b2.x = Bp[32];
    b2.y = Bp[(size_t)N + 32];
    b3.x = Bp[48];
    b3.y = Bp[(size_t)N + 48];
    acc0 = __builtin_amdgcn_wmma_f32_16x16x4_f32(false, a, false, b0, (short)0,
                                                 acc0, false, false);
    acc1 = __builtin_amdgcn_wmma_f32_16x16x4_f32(false, a, false, b1, (short)0,
                                                 acc1, false, false);
    acc2 = __builtin_amdgcn_wmma_f32_16x16x4_f32(false, a, false, b2, (short)0,
                                                 acc2, false, false);
    acc3 = __builtin_amdgcn_wmma_f32_16x16x4_f32(false, a, false, b3, (short)0,
                                                 acc3, false, false);
    Ap += 4;
    Bp += bstep;
  }
  const int mb = m0 + ((lane >> 4) << 3);
  v8f accs[4] = {acc0, acc1, acc2, acc3};
  float p = 0.f;
#pragma unroll
  for (int t = 0; t < 4; ++t) {
    const int n = n0 + 16 * t + mh;
    const float qn = q[n];
    const float kbn = kb[n];
#pragma unroll
    for (int j = 0; j < 8; ++j) {
      int m = mb + j;
      if (m < M) p += qn * tanhf(accs[t][j] + kbn);
    }
  }
#pragma unroll
  for (int off = 16; off > 0; off >>= 1) p += __shfl_down(p, off, 32);
  if (lane == 0) atomicAdd(score, p);
}

// ---------------------------------------------------------------------------
// Fill / elementwise
// ---------------------------------------------------------------------------
__global__ void fill_f32(float* __restrict__ p, float v, size_t n) {
  size_t t = (size_t)blockIdx.x * blockDim.x + threadIdx.x;
  if (t < n) p[t] = v;
}
__global__ void fill_u32(unsigned* __restrict__ p, unsigned v, size_t n) {
  size_t t = (size_t)blockIdx.x * blockDim.x + threadIdx.x;
  if (t < n) p[t] = v;
}
__global__ void relu_inplace(float* __restrict__ p, size_t n) {
  size_t t = (size_t)blockIdx.x * blockDim.x + threadIdx.x;
  if (t < n) p[t] = fmaxf(p[t], 0.f);
}

// ---------------------------------------------------------------------------
// a[n,h] = sum_d h[n, h*D+d] * att[h*D+d]
// ---------------------------------------------------------------------------
__global__ void att_proj(const float* __restrict__ h,
                         const float* __restrict__ att,
                         float* __restrict__ out, int Nn, int H, int D) {
  size_t t = (size_t)blockIdx.x * blockDim.x + threadIdx.x;
  if (t >= (size_t)Nn * H) return;
  int n = (int)(t / H), hh = (int)(t % H);
  const float* hr = h + (size_t)n * H * D + (size_t)hh * D;
  const float* ar = att + hh * D;
  float s = 0.f;
  for (int d = 0; d < D; ++d) s += hr[d] * ar[d];
  out[t] = s;
}

// alpha = lrelu(a_src[row]+a_dst[col], 0.1); segment max (ordered-uint atomic)
__global__ void edge_alpha_max(const int* __restrict__ row,
                               const int* __restrict__ col,
                               const float* __restrict__ as,
                               const float* __restrict__ ad,
                               float* __restrict__ alpha,
                               unsigned* __restrict__ mkey, int E, int H) {
  size_t t = (size_t)blockIdx.x * blockDim.x + threadIdx.x;
  if (t >= (size_t)E * H) return;
  int e = (int)(t / H), hh = (int)(t % H);
  float v = as[(size_t)row[e] * H + hh] + ad[(size_t)col[e] * H + hh];
  v = (v >= 0.f) ? v : 0.1f * v;
  alpha[t] = v;
  atomicMax(&mkey[(size_t)col[e] * H + hh], fenc(v));
}

// alpha <- exp(alpha - m[col]); segment sum
__global__ void edge_expsum(const int* __restrict__ col,
                            float* __restrict__ alpha,
                            const unsigned* __restrict__ mkey,
                            float* __restrict__ ssum, int E, int H) {
  size_t t = (size_t)blockIdx.x * blockDim.x + threadIdx.x;
  if (t >= (size_t)E * H) return;
  int e = (int)(t / H), hh = (int)(t % H);
  size_t ci = (size_t)col[e] * H + hh;
  float m = fdec(mkey[ci]);
  if (!__builtin_isfinite(m)) m = 0.f;  // mirrors reference isfinite guard
  float ex = expf(alpha[t] - m);
  alpha[t] = ex;
  atomicAdd(&ssum[ci], ex);
}

// out[col] += h_src[row] * (alpha / (ssum[col]+eps)); one wave per edge
__global__ __launch_bounds__(256) void edge_msg(
    const int* __restrict__ row, const int* __restrict__ col,
    const float* __restrict__ hsrc, const float* __restrict__ alpha,
    const float* __restrict__ ssum, float* __restrict__ out, int E, int H,
    int D) {
  int e = blockIdx.x * (blockDim.x >> 5) + (threadIdx.x >> 5);
  if (e >= E) return;
  int lane = threadIdx.x & 31;
  int r = row[e], c = col[e];
  int C = H * D;
  for (int ch = lane; ch < C; ch += 32) {
    int hh = ch / D;
    float w = alpha[(size_t)e * H + hh] /
              (ssum[(size_t)c * H + hh] + SEG_EPS);
    atomicAdd(&out[(size_t)c * C + ch], hsrc[(size_t)r * C + ch] * w);
  }
}

// res = elu( sum_r softmax_r(score*invN) * rel[r] )
__global__ void combine_elu(const float* __restrict__ rel, size_t slotStride,
                            int R, const float* __restrict__ score, float invN,
                            float* __restrict__ res, size_t total) {
  size_t t = (size_t)blockIdx.x * blockDim.x + threadIdx.x;
  if (t >= total) return;
  float mx = -3.4e38f;
  for (int r = 0; r < R; ++r) mx = fmaxf(mx, score[r] * invN);
  float den = 0.f, acc = 0.f;
  for (int r = 0; r < R; ++r) {
    float w = expf(score[r] * invN - mx);
    den += w;
    acc += w * rel[(size_t)r * slotStride + t];
  }
  float v = acc / den;
  res[t] = (v > 0.f) ? v : expm1f(v);
}

// ---------------------------------------------------------------------------
// Set transformer (m=8 inducing points, 4 heads, dh=128, dm=512)
// ---------------------------------------------------------------------------
__global__ void st_scores(const float* __restrict__ qp,
                          const float* __restrict__ kp,
                          float* __restrict__ sc, int N) {
  size_t t = (size_t)blockIdx.x * blockDim.x + threadIdx.x;
  if (t >= (size_t)N * 32) return;
  int n = (int)(t >> 5);
  int hm = (int)(t & 31);
  int h = hm >> 3, m = hm & 7;
  const float* qr = qp + m * 512 + h * 128;
  const float* kr = kp + (size_t)n * 512 + h * 128;
  float s = 0.f;
  for (int d = 0; d < 128; ++d) s += qr[d] * kr[d];
  sc[(size_t)hm * N + n] = s * 0.08838834764831845f;  // 1/sqrt(128)
}

__global__ __launch_bounds__(256) void st_stats(const float* __restrict__ sc,
                                                float* __restrict__ smax,
                                                float* __restrict__ ssum,
                                                int N) {
  __shared__ float red[256];
  int hm = blockIdx.x;
  const float* s = sc + (size_t)hm * N;
  float mx = -3.4e38f;
  for (int n = threadIdx.x; n < N; n += 256) mx = fmaxf(mx, s[n]);
  red[threadIdx.x] = mx;
  __syncthreads();
  for (int o = 128; o > 0; o >>= 1) {
    if (threadIdx.x < o)
      red[threadIdx.x] = fmaxf(red[threadIdx.x], red[threadIdx.x + o]);
    __syncthreads();
  }
  float m = red[0];
  __syncthreads();
  float sm = 0.f;
  for (int n = threadIdx.x; n < N; n += 256) sm += expf(s[n] - m);
  red[threadIdx.x] = sm;
  __syncthreads();
  for (int o = 128; o > 0; o >>= 1) {
    if (threadIdx.x < o) red[threadIdx.x] += red[threadIdx.x + o];
    __syncthreads();
  }
  if (threadIdx.x == 0) {
    smax[hm] = m;
    ssum[hm] = red[0];
  }
}

__global__ __launch_bounds__(128) void st_attn_out(
    const float* __restrict__ sc, const float* __restrict__ smax,
    const float* __restrict__ ssum, const float* __restrict__ vp,
    float* __restrict__ o, int N) {
  __shared__ float wsh[128];
  int hm = blockIdx.x;
  int h = hm >> 3, m = hm & 7;
  int d = threadIdx.x;
  float mx = smax[hm];
  float inv = 1.f / ssum[hm];
  const float* s = sc + (size_t)hm * N;
  float acc = 0.f;
  for (int nb = 0; nb < N; nb += 128) {
    int n = nb + threadIdx.x;
    wsh[threadIdx.x] = (n < N) ? expf(s[n] - mx) : 0.f;
    __syncthreads();
    int lim = (N - nb < 128) ? (N - nb) : 128;
    for (int i = 0; i < lim; ++i)
      acc += wsh[i] * vp[(size_t)(nb + i) * 512 + h * 128 + d];
    __syncthreads();
  }
  o[m * 512 + h * 128 + d] = acc * inv;
}

__global__ void st_mean(const float* __restrict__ o2, float* __restrict__ mo) {
  int d = blockIdx.x * blockDim.x + threadIdx.x;
  if (d >= 512) return;
  float s = 0.f;
  for (int m = 0; m < 8; ++m) s += o2[m * 512 + d];
  mo[d] = s * 0.125f;
}

__global__ void vecmat(const float* __restrict__ x, const float* __restrict__ W,
                       const float* __restrict__ b, float* __restrict__ y,
                       int K, int N) {
  int n = blockIdx.x * blockDim.x + threadIdx.x;
  if (n >= N) return;
  float s = b[n];
  for (int k = 0; k < K; ++k) s += x[k] * W[(size_t)k * N + n];
  y[n] = s;
}

__global__ void add_lrelu(const float* __restrict__ a,
                          const float* __restrict__ b, float* __restrict__ y,
                          int n) {
  int t = blockIdx.x * blockDim.x + threadIdx.x;
  if (t >= n) return;
  float v = a[t] + b[t];
  y[t] = (v >= 0.f) ? v : 0.01f * v;
}

// ---------------------------------------------------------------------------
// Host driver
// ---------------------------------------------------------------------------
static inline int cdiv(long long a, long long b) { return (int)((a + b - 1) / b); }

extern "C" void kernel_launch(void* const* d_in, const int* in_sizes, int n_in,
                              void* d_out, int out_size, void* d_ws,
                              size_t ws_size, hipStream_t stream) {
  (void)ws_size; (void)out_size;
  if (n_in < 81) return;
  auto F = [&](int i) { return (const float*)d_in[i]; };
  auto Ied = [&](int i) { return (const int*)d_in[i]; };

  const int NI = in_sizes[0] / 128;
  const int NV = in_sizes[1] / 64;
  const int NC = in_sizes[2] / 32;
  const int Ec = in_sizes[3] / 2;   // control
  const int Ecl = in_sizes[4] / 2;  // call
  const int Eiv = in_sizes[5] / 2;  // iv
  const int Evi = in_sizes[6] / 2;  // vi
  const int Eci = in_sizes[7] / 2;  // ci

  // ---- workspace carve (bytes, 256-aligned) ----
  char* base = (char*)d_ws;
  size_t off = 0;
  auto carve = [&](size_t bytes) -> float* {
    float* p = (float*)(base + off);
    off += (bytes + 255) & ~(size_t)255;
    return p;
  };
  float* xi = carve((size_t)NI * 512 * 4);
  float* xv = carve((size_t)NV * 512 * 4);
  float* hi = carve((size_t)NI * 512 * 4);
  float* hv = carve((size_t)NV * 512 * 4);
  float* hc = carve((size_t)NC * 256 * 4);
  size_t slotStride = (size_t)NI * 512;  // floats
  float* slots = carve(4 * slotStride * 4);
  float* slotv = carve((size_t)NV * 512 * 4);
  float* asrc = carve((size_t)NI * 8 * 4);
  float* adst = carve((size_t)NI * 8 * 4);
  float* alphab = carve((size_t)Ec * 8 * 4);
  unsigned* mkey = (unsigned*)carve((size_t)NI * 8 * 4);
  float* ssum = carve((size_t)NI * 8 * 4);
  float* scoreI = carve(4 * 4);
  float* scoreV = carve(4 * 4);
  float* qp = carve(8 * 512 * 4);
  float* stsc = carve((size_t)32 * NI * 4);
  float* stmax = carve(32 * 4);
  float* stsum = carve(32 * 4);
  float* obuf = carve(8 * 512 * 4);
  float* o2buf = carve(8 * 512 * 4);
  float* meano = carve(512 * 4);
  float* stoutI = carve(256 * 4);
  float* stoutV = carve(256 * 4);
  float* aggb = carve(256 * 4);

  // ---- launch helpers ----
  auto gemm = [&](const float* A, const float* B, const float* bias, float* C,
                  int M, int N, int K, int act) {
    dim3 g(N / 64, cdiv(M, 64)), b(32, 4);
    gemm_wmma<<<g, b, 0, stream>>>(A, B, bias, C, M, N, K, act);
  };
  auto gemmScore = [&](const float* A, const float* B, const float* kb,
                       const float* q, float* sc, int M, int N, int K) {
    dim3 g(N / 64, cdiv(M, 64)), b(32, 4);
    gemm_wmma_score<<<g, b, 0, stream>>>(A, B, kb, q, sc, M, N, K);
  };
  auto fillF = [&](float* p, float v, size_t n) {
    fill_f32<<<cdiv((long long)n, 256), 256, 0, stream>>>(p, v, n);
  };
  auto fillU = [&](unsigned* p, unsigned v, size_t n) {
    fill_u32<<<cdiv((long long)n, 256), 256, 0, stream>>>(p, v, n);
  };

  auto doRel = [&](const int* epair, int E, const float* hsrc, int nsrc,
                   const float* hdst, int ndst, const float* attS,
                   const float* attD, float* outSlot, int H, int D) {
    const int C = H * D;
    const int* row = epair;
    const int* colp = epair + E;
    att_proj<<<cdiv((long long)nsrc * H, 256), 256, 0, stream>>>(hsrc, attS,
                                                                 asrc, nsrc, H, D);
    att_proj<<<cdiv((long long)ndst * H, 256), 256, 0, stream>>>(hdst, attD,
                                                                 adst, ndst, H, D);
    fillU(mkey, 0x007FFFFFu, (size_t)ndst * H);  // enc(-inf)
    fillF(ssum, 0.f, (size_t)ndst * H);
    edge_alpha_max<<<cdiv((long long)E * H, 256), 256, 0, stream>>>(
        row, colp, asrc, adst, alphab, mkey, E, H);
    edge_expsum<<<cdiv((long long)E * H, 256), 256, 0, stream>>>(
        colp, alphab, mkey, ssum, E, H);
    fillF(outSlot, 0.f, (size_t)ndst * C);
    edge_msg<<<cdiv(E, 8), 256, 0, stream>>>(row, colp, hsrc, alphab, ssum,
                                             outSlot, E, H, D);
    relu_inplace<<<cdiv((long long)ndst * C, 256), 256, 0, stream>>>(
        outSlot, (size_t)ndst * C);
  };

  // ============================ Layer 1 (H=8,D=32,C=256) ====================
  gemm(F(0), F(24), F(25), hi, NI, 256, 128, 0);  // lin.inst
  gemm(F(1), F(26), F(27), hv, NV, 256, 64, 0);   // lin.var
  gemm(F(2), F(22), F(23), hc, NC, 256, 32, 0);   // lin.const
  fillF(scoreI, 0.f, 4);
  fillF(scoreV, 0.f, 4);
  {
    const float *kW = F(20), *kb = F(21), *q = F(28);
    // control: inst->inst  (slot 0)
    doRel(Ied(3), Ec, hi, NI, hi, NI, F(17), F(12), slots + 0 * slotStride, 8, 32);
    gemmScore(slots + 0 * slotStride, kW, kb, q, scoreI + 0, NI, 256, 256);
    // call: inst->inst  (slot 1)
    doRel(Ied(4), Ecl, hi, NI, hi, NI, F(15), F(10), slots + 1 * slotStride, 8, 32);
    gemmScore(slots + 1 * slotStride, kW, kb, q, scoreI + 1, NI, 256, 256);
    // iv: inst->var  (var slot)
    doRel(Ied(5), Eiv, hi, NI, hv, NV, F(18), F(13), slotv, 8, 32);
    gemmScore(slotv, kW, kb, q, scoreV + 0, NV, 256, 256);
    // vi: var->inst  (slot 2)
    doRel(Ied(6), Evi, hv, NV, hi, NI, F(19), F(14), slots + 2 * slotStride, 8, 32);
    gemmScore(slots + 2 * slotStride, kW, kb, q, scoreI + 2, NI, 256, 256);
    // ci: const->inst  (slot 3)
    doRel(Ied(7), Eci, hc, NC, hi, NI, F(16), F(11), slots + 3 * slotStride, 8, 32);
    gemmScore(slots + 3 * slotStride, kW, kb, q, scoreI + 3, NI, 256, 256);
    combine_elu<<<cdiv((long long)NI * 256, 256), 256, 0, stream>>>(
        slots, slotStride, 4, scoreI, 1.f / NI, xi, (size_t)NI * 256);
    combine_elu<<<cdiv((long long)NV * 256, 256), 256, 0, stream>>>(
        slotv, slotStride, 1, scoreV, 1.f / NV, xv, (size_t)NV * 256);
  }

  // ============================ Layer 2 (H=8,D=32,C=256) ====================
  gemm(xi, F(39), F(40), hi, NI, 256, 256, 0);
  gemm(xv, F(41), F(42), hv, NV, 256, 256, 0);
  fillF(scoreI, 0.f, 4);
  fillF(scoreV, 0.f, 4);
  {
    const float *kW = F(37), *kb = F(38), *q = F(43);
    doRel(Ied(3), Ec, hi, NI, hi, NI, F(34), F(30), slots + 0 * slotStride, 8, 32);
    gemmScore(slots + 0 * slotStride, kW, kb, q, scoreI + 0, NI, 256, 256);
    doRel(Ied(4), Ecl, hi, NI, hi, NI, F(33), F(29), slots + 1 * slotStride, 8, 32);
    gemmScore(slots + 1 * slotStride, kW, kb, q, scoreI + 1, NI, 256, 256);
    doRel(Ied(5), Eiv, hi, NI, hv, NV, F(35), F(31), slotv, 8, 32);
    gemmScore(slotv, kW, kb, q, scoreV + 0, NV, 256, 256);
    doRel(Ied(6), Evi, hv, NV, hi, NI, F(36), F(32), slots + 2 * slotStride, 8, 32);
    gemmScore(slots + 2 * slotStride, kW, kb, q, scoreI + 2, NI, 256, 256);
    combine_elu<<<cdiv((long long)NI * 256, 256), 256, 0, stream>>>(
        slots, slotStride, 3, scoreI, 1.f / NI, xi, (size_t)NI * 256);
    combine_elu<<<cdiv((long long)NV * 256, 256), 256, 0, stream>>>(
        slotv, slotStride, 1, scoreV, 1.f / NV, xv, (size_t)NV * 256);
  }

  // ============================ Layer 3 (H=4,D=128,C=512) ===================
  gemm(xi, F(54), F(55), hi, NI, 512, 256, 0);
  gemm(xv, F(56), F(57), hv, NV, 512, 256, 0);
  fillF(scoreI, 0.f, 4);
  fillF(scoreV, 0.f, 4);
  {
    const float *kW = F(52), *kb = F(53), *q = F(58);
    doRel(Ied(3), Ec, hi, NI, hi, NI, F(49), F(45), slots + 0 * slotStride, 4, 128);
    gemmScore(slots + 0 * slotStride, kW, kb, q, scoreI + 0, NI, 512, 512);
    doRel(Ied(4), Ecl, hi, NI, hi, NI, F(48), F(44), slots + 1 * slotStride, 4, 128);
    gemmScore(slots + 1 * slotStride, kW, kb, q, scoreI + 1, NI, 512, 512);
    doRel(Ied(5), Eiv, hi, NI, hv, NV, F(50), F(46), slotv, 4, 128);
    gemmScore(slotv, kW, kb, q, scoreV + 0, NV, 512, 512);
    doRel(Ied(6), Evi, hv, NV, hi, NI, F(51), F(47), slots + 2 * slotStride, 4, 128);
    gemmScore(slots + 2 * slotStride, kW, kb, q, scoreI + 2, NI, 512, 512);
    combine_elu<<<cdiv((long long)NI * 512, 256), 256, 0, stream>>>(
        slots, slotStride, 3, scoreI, 1.f / NI, xi, (size_t)NI * 512);
    combine_elu<<<cdiv((long long)NV * 512, 256), 256, 0, stream>>>(
        slotv, slotStride, 1, scoreV, 1.f / NV, xv, (size_t)NV * 512);
  }

  // ============================ Set transformers ============================
  auto runST = [&](const float* x, int N, int pb, float* stout) {
    const float *Wf = F(pb + 0), *Wk = F(pb + 1), *Wo = F(pb + 2),
                *Wq = F(pb + 3), *Wv = F(pb + 4);
    const float *bf = F(pb + 5), *bk = F(pb + 6), *bo = F(pb + 7),
                *bq = F(pb + 8), *bv = F(pb + 9), *ind = F(pb + 10);
    float* kp = slots;                 // reuse freed relation slots
    float* vp = slots + slotStride;
    gemm(ind, Wq, bq, qp, 8, 512, 512, 0);
    gemm(x, Wk, bk, kp, N, 512, 512, 0);
    gemm(x, Wv, bv, vp, N, 512, 512, 0);
    st_scores<<<cdiv((long long)N * 32, 256), 256, 0, stream>>>(qp, kp, stsc, N);
    st_stats<<<32, 256, 0, stream>>>(stsc, stmax, stsum, N);
    st_attn_out<<<32, 128, 0, stream>>>(stsc, stmax, stsum, vp, obuf, N);
    gemm(obuf, Wo, bo, o2buf, 8, 512, 512, 0);
    st_mean<<<2, 256, 0, stream>>>(o2buf, meano);
    vecmat<<<1, 256, 0, stream>>>(meano, Wf, bf, stout, 512, 256);
  };
  runST(xi, NI, 59, stoutI);
  runST(xv, NV, 70, stoutV);

  add_lrelu<<<1, 256, 0, stream>>>(stoutI, stoutV, aggb, 256);
  vecmat<<<1, 64, 0, stream>>>(aggb, F(8), F(9), (float*)d_out, 256, 64);
}